// DFPS_Sampler_54898271977791
// MI455X (gfx1250) — compile-verified
//
#include <hip/hip_runtime.h>

// Density + Manhattan-weighted FPS for MI455X (gfx1250, wave32).
//
// Pass 1 (density): pairwise d^2 via V_WMMA_F32_16X16X4_F32.
//   A[m]   = ( x_m,  y_m,  z_m, |p_m|^2 )        (K = 4)
//   B[:,n] = (-2x_n, -2y_n, -2z_n, 1 )
//   C[m][n] = |p_n|^2  (per-lane broadcast: lane == fixed column in C/D layout)
//   => D[m][n] = |p_m|^2 + |p_n|^2 - 2 p_m.p_n = d^2(m,n), straight out of the MAC unit.
// One wave owns a 16-column tile, loops over 512 row tiles, counts d^2 <= R^2
// per column in registers (no atomics), halves combined with shfl_xor(16).
//
// Pass 2 (FPS): one 1024-thread block per batch; 8 points/thread in VGPRs,
// argmax via wave shfl reduction + 32-entry LDS reduction; winning thread
// publishes the selected point's coords from its registers to LDS.

typedef __attribute__((ext_vector_type(2))) float v2f;
typedef __attribute__((ext_vector_type(8))) float v8f;

#define NB      4
#define NPTS    8192
#define NPOINT  2048
#define RAD2    0.16f           // R*R, R = 0.4
#define FPS_T   1024            // threads per FPS block
#define K_SLOT  (NPTS / FPS_T)  // 8 points per thread

__global__ __launch_bounds__(256)
void dfps_density_wmma(const float* __restrict__ pts, float* __restrict__ pen) {
  const int lane = threadIdx.x & 31;
  const int gw   = blockIdx.x * (blockDim.x >> 5) + (threadIdx.x >> 5);
  const int b    = gw >> 9;            // 512 column tiles per batch
  const int j    = gw & 511;
  const float* __restrict__ P = pts + (size_t)b * NPTS * 3;

  // ---- B fragment + C (fixed for this wave's 16-column tile) ----
  const int   nc   = (j << 4) + (lane & 15);
  const float bx   = P[nc * 3 + 0];
  const float by   = P[nc * 3 + 1];
  const float bz   = P[nc * 3 + 2];
  const float nrmB = bx * bx + by * by + bz * bz;

  v2f fb;
  // B layout (mirrors ISA 16x4 A layout): lanes 0-15 hold K0,K1; lanes 16-31 hold K2,K3
  fb.x = (lane < 16) ? (-2.0f * bx) : (-2.0f * bz);
  fb.y = (lane < 16) ? (-2.0f * by) : 1.0f;

  v8f c;
#pragma unroll
  for (int r = 0; r < 8; ++r) c[r] = nrmB;   // C[m][n] = |p_n|^2 (lane = column)

  float cnt = 0.0f;

  for (int i = 0; i < NPTS / 16; ++i) {
    const int m = (i << 4) + (lane & 15);
    const float* __restrict__ q = P + m * 3;
    __builtin_prefetch(q + 16 * 3, 0, 1);            // global_prefetch_b8: next row tile
    const float ax = q[0], ay = q[1], az = q[2];
    const float nrmA = ax * ax + ay * ay + az * az;

    v2f fa;
    // A 16x4 layout: lanes 0-15 -> (K0,K1) = (x,y); lanes 16-31 -> (K2,K3) = (z,|p|^2)
    fa.x = (lane < 16) ? ax : az;
    fa.y = (lane < 16) ? ay : nrmA;

    // D = A*B + C  ==  d^2 tile (16x16), f32 throughout.
    v8f d = __builtin_amdgcn_wmma_f32_16x16x4_f32(
        /*neg_a=*/false, fa, /*neg_b=*/false, fb,
        /*c_mod=*/(short)0, c, /*reuse_a=*/false, /*reuse_b=*/false);

#pragma unroll
    for (int r = 0; r < 8; ++r) cnt += (d[r] <= RAD2) ? 1.0f : 0.0f;
  }

  // lane l counts rows m%16 in 0..7, lane l+16 counts rows 8..15 (same column)
  const float other = __shfl_xor(cnt, 16, 32);
  if (lane < 16) pen[(size_t)b * NPTS + nc] = 1.0f / (cnt + other);   // density^-ALPHA
}

__global__ __launch_bounds__(FPS_T)
void dfps_fps(const float* __restrict__ pts, const float* __restrict__ pen,
              int* __restrict__ out) {
  const int b    = blockIdx.x;
  const int tid  = threadIdx.x;
  const int lane = tid & 31;
  const int wv   = tid >> 5;
  const float* __restrict__ P = pts + (size_t)b * NPTS * 3;
  const float* __restrict__ W = pen + (size_t)b * NPTS;
  int* __restrict__ O = out + (size_t)b * NPOINT;

  float px[K_SLOT], py[K_SLOT], pz[K_SLOT], pw[K_SLOT], md[K_SLOT];
#pragma unroll
  for (int k = 0; k < K_SLOT; ++k) {
    const int p = tid + k * FPS_T;
    px[k] = P[p * 3 + 0];
    py[k] = P[p * 3 + 1];
    pz[k] = P[p * 3 + 2];
    pw[k] = W[p];
    md[k] = 1e10f;
  }

  __shared__ float redV[32];
  __shared__ int   redI[32];
  __shared__ int   shCur;
  __shared__ float shC[3];

  if (tid == 0) {                 // scan emits initial cur = 0 first
    O[0]   = 0;
    shCur  = 0;
    shC[0] = px[0]; shC[1] = py[0]; shC[2] = pz[0];   // thread 0 owns point 0
  }
  __syncthreads();

  for (int t = 1; t < NPOINT; ++t) {
    const float cx = shC[0], cy = shC[1], cz = shC[2];

    float bv = -1.0f;
    int   bi = 0x7fffffff;
#pragma unroll
    for (int k = 0; k < K_SLOT; ++k) {
      const float d = fabsf(px[k] - cx) + fabsf(py[k] - cy) + 2.0f * fabsf(pz[k] - cz);
      md[k] = fminf(md[k], d);
      const float v   = md[k] * pw[k];
      const int   idx = tid + k * FPS_T;
      if (v > bv || (v == bv && idx < bi)) { bv = v; bi = idx; }   // first-occurrence argmax
    }

    // wave32 reduction
#pragma unroll
    for (int off = 16; off > 0; off >>= 1) {
      const float ov = __shfl_down(bv, off, 32);
      const int   oi = __shfl_down(bi, off, 32);
      if (ov > bv || (ov == bv && oi < bi)) { bv = ov; bi = oi; }
    }
    if (lane == 0) { redV[wv] = bv; redI[wv] = bi; }
    __syncthreads();

    if (wv == 0) {                // 32 waves -> wave 0 finishes the block argmax
      bv = redV[lane];
      bi = redI[lane];
#pragma unroll
      for (int off = 16; off > 0; off >>= 1) {
        const float ov = __shfl_down(bv, off, 32);
        const int   oi = __shfl_down(bi, off, 32);
        if (ov > bv || (ov == bv && oi < bi)) { bv = ov; bi = oi; }
      }
      if (lane == 0) { shCur = bi; O[t] = bi; }
    }
    __syncthreads();

    const int cur = shCur;
    if ((cur & (FPS_T - 1)) == tid) {       // owner publishes coords from registers
      const int k = cur >> 10;
      shC[0] = px[k]; shC[1] = py[k]; shC[2] = pz[k];
    }
    __syncthreads();
  }
}

extern "C" void kernel_launch(void* const* d_in, const int* in_sizes, int n_in,
                              void* d_out, int out_size, void* d_ws, size_t ws_size,
                              hipStream_t stream) {
  (void)in_sizes; (void)n_in; (void)out_size; (void)ws_size;
  const float* points = (const float*)d_in[0];   // [B, N, 3] f32
  // d_in[1] = features (unused by the sampler), d_in[2] = npoint scalar (fixed 2048)
  float* pen = (float*)d_ws;                     // [B, N] f32 penalty, 128 KB scratch
  int*   out = (int*)d_out;                      // [B, npoint] int32 indices

  // 4 batches * 512 column tiles = 2048 waves; 8 waves (256 thr) per block.
  dfps_density_wmma<<<(NB * (NPTS / 16)) / 8, 256, 0, stream>>>(points, pen);
  dfps_fps<<<NB, FPS_T, 0, stream>>>(points, pen, out);
}